// CF_34016140984494
// MI455X (gfx1250) — compile-verified
//
#include <hip/hip_runtime.h>
#include <hip/hip_bf16.h>
#include <stdint.h>

// ---------------------------------------------------------------------------
// CF block on gfx1250: fp32 WMMA (V_WMMA_F32_16X16X4_F32) for both big GEMMs,
// LDS-staged operands (async global->LDS), double-buffered sampled tiles for
// the deformable conv.
// ---------------------------------------------------------------------------

typedef float v2f __attribute__((ext_vector_type(2)));
typedef float v8f __attribute__((ext_vector_type(8)));

#define NB   8
#define CIN  512
#define MID  256
#define OUTC 256
#define HH   56
#define WW   56
#define NPIX (HH * WW)      // 3136 = 98 * 32
#define HY   28
#define WY   28
#define KPTS 9
#define EPSB 1e-5f

// -------------------- async global->LDS helpers (gfx1250) -------------------
#if defined(__has_builtin)
#  if __has_builtin(__builtin_amdgcn_global_load_async_to_lds_b32)
#    define HAVE_ASYNC_BUILTIN 1
#  endif
#  if __has_builtin(__builtin_amdgcn_s_wait_asynccnt)
#    define HAVE_WAIT_ASYNC_BUILTIN 1
#  endif
#endif

typedef __attribute__((address_space(1))) int as1_int;
typedef __attribute__((address_space(3))) int as3_int;

__device__ __forceinline__ void async_g2l_b32(float* lds_dst, const float* gsrc) {
#if defined(HAVE_ASYNC_BUILTIN)
    __builtin_amdgcn_global_load_async_to_lds_b32(
        (as1_int*)gsrc, (as3_int*)lds_dst, 0, 0);
#else
    asm volatile("global_load_async_to_lds_b32 %0, %1, off"
                 :: "v"((uint32_t)(uintptr_t)lds_dst), "v"(gsrc) : "memory");
#endif
}

__device__ __forceinline__ void async_wait0() {
#if defined(HAVE_WAIT_ASYNC_BUILTIN)
    __builtin_amdgcn_s_wait_asynccnt(0);
#else
    asm volatile("s_wait_asynccnt 0x0" ::: "memory");
#endif
}

// ---------------------------------------------------------------------------
// Kernel 1: xm = BN1(x @ conv1_w^T); yu = nearest-up(y);
//           z = sigmoid(att_w @ cat(xm,yu) + att_b); xf = xm*z0 + yu*z1
// Block = 512 thr = 16 waves; block covers 256 mids x 32 pixels.
// x tile (512 x 32) staged in LDS via async copies; each wave: 2 chained-WMMA
// streams (pixels 0..15 / 16..31) sharing A fragments.
// ---------------------------------------------------------------------------
__global__ __launch_bounds__(512) void cf_fuse1(
    const float* __restrict__ x,    const float* __restrict__ y,
    const float* __restrict__ w1,
    const float* __restrict__ g1,   const float* __restrict__ b1,
    const float* __restrict__ mu1,  const float* __restrict__ var1,
    const float* __restrict__ attw, const float* __restrict__ attb,
    float* __restrict__ xf)
{
    __shared__ float xlds[32][CIN + 4];   // +4 dwords pad: conflict-free ds_b64
    __shared__ float zsum[2][32];

    const int tid  = threadIdx.x;
    const int wave = tid >> 5;            // 0..15 -> mid tile
    const int lane = tid & 31;
    const int nn   = lane & 15;
    const int hl   = lane >> 4;
    const int nb   = blockIdx.x / (NPIX / 32);
    const int p0   = (blockIdx.x % (NPIX / 32)) * 32;
    const int m0   = wave * 16;

    if (tid < 64) zsum[tid >> 5][tid & 31] = 0.0f;

    // ---- stage x[k, p0..p0+31] into LDS (async global->LDS) ----
    const float* xp = x + (size_t)nb * CIN * NPIX;
    {
        const int pix = tid & 31;
        const int kb0 = (tid >> 5) * 32;              // 16 groups of 32 k
        const float* src = xp + (size_t)kb0 * NPIX + p0 + pix;
        float* dst = &xlds[pix][kb0];
#pragma unroll 8
        for (int j = 0; j < 32; ++j)
            async_g2l_b32(dst + j, src + (size_t)j * NPIX);
        async_wait0();
    }
    __syncthreads();

    // ---- GEMM: 128 chained wmma per accumulator, A shared by both ----
    const float* wrow  = w1 + (size_t)(m0 + nn) * CIN;
    const float* brow0 = &xlds[nn][0];
    const float* brow1 = &xlds[nn + 16][0];
    v8f acc0 = {}, acc1 = {};
    for (int kb = 0; kb < CIN; kb += 4) {
        const int k = kb + 2 * hl;
        v2f a  = { wrow[k],  wrow[k + 1]  };
        v2f b0 = { brow0[k], brow0[k + 1] };
        v2f b1 = { brow1[k], brow1[k + 1] };
        acc0 = __builtin_amdgcn_wmma_f32_16x16x4_f32(
            false, a, false, b0, (short)0, acc0, false, false);
        acc1 = __builtin_amdgcn_wmma_f32_16x16x4_f32(
            false, a, false, b1, (short)0, acc1, false, false);
    }

    // ---- BN1 + upsampled y + attention partial sums (both pixel sets) ----
    const int pA = p0 + nn, pB = p0 + nn + 16;
    const int hA = pA / WW, wA = pA % WW;
    const int hB = pB / WW, wB = pB % WW;
    const float* ybase = y + (size_t)nb * MID * (HY * WY);
    const float* ypA = ybase + (size_t)(hA >> 1) * WY + (wA >> 1);
    const float* ypB = ybase + (size_t)(hB >> 1) * WY + (wB >> 1);

    float xmv[2][8], yv[2][8];
    float s0A = 0.f, s1A = 0.f, s0B = 0.f, s1B = 0.f;
#pragma unroll
    for (int r = 0; r < 8; ++r) {
        const int m  = m0 + r + 8 * hl;
        const float sc = g1[m] * rsqrtf(var1[m] + EPSB);
        const float sh = b1[m] - mu1[m] * sc;
        const float a0 = attw[m], a1 = attw[256 + m];
        const float a2 = attw[512 + m], a3 = attw[768 + m];

        const float xvA = acc0[r] * sc + sh;
        const float yyA = ypA[(size_t)m * (HY * WY)];
        xmv[0][r] = xvA; yv[0][r] = yyA;
        s0A += a0 * xvA + a1 * yyA;
        s1A += a2 * xvA + a3 * yyA;

        const float xvB = acc1[r] * sc + sh;
        const float yyB = ypB[(size_t)m * (HY * WY)];
        xmv[1][r] = xvB; yv[1][r] = yyB;
        s0B += a0 * xvB + a1 * yyB;
        s1B += a2 * xvB + a3 * yyB;
    }
    atomicAdd(&zsum[0][nn], s0A);      atomicAdd(&zsum[1][nn], s1A);
    atomicAdd(&zsum[0][nn + 16], s0B); atomicAdd(&zsum[1][nn + 16], s1B);
    __syncthreads();

    const float sg0A = 1.0f / (1.0f + __expf(-(zsum[0][nn] + attb[0])));
    const float sg1A = 1.0f / (1.0f + __expf(-(zsum[1][nn] + attb[1])));
    const float sg0B = 1.0f / (1.0f + __expf(-(zsum[0][nn + 16] + attb[0])));
    const float sg1B = 1.0f / (1.0f + __expf(-(zsum[1][nn + 16] + attb[1])));

    float* xfp = xf + (size_t)nb * MID * NPIX;
#pragma unroll
    for (int r = 0; r < 8; ++r) {
        const int m = m0 + r + 8 * hl;
        xfp[(size_t)m * NPIX + pA] = xmv[0][r] * sg0A + yv[0][r] * sg1A;
        xfp[(size_t)m * NPIX + pB] = xmv[1][r] * sg0B + yv[1][r] * sg1B;
    }
}

// ---------------------------------------------------------------------------
// Kernel 2: offset = conv3x3(xf, off_w), 18 output channels (direct, ~5% FLOPs)
// ---------------------------------------------------------------------------
__global__ __launch_bounds__(256) void cf_offconv(
    const float* __restrict__ xf, const float* __restrict__ offw,
    float* __restrict__ off)
{
    const int idx = blockIdx.x * 256 + threadIdx.x;
    if (idx >= NB * 18 * NPIX) return;
    const int p  = idx % NPIX;
    const int t  = idx / NPIX;
    const int o  = t % 18;
    const int nb = t / 18;
    const int h  = p / WW, w_ = p % WW;

    const float* xp = xf   + (size_t)nb * MID * NPIX;
    const float* wp = offw + (size_t)o * MID * 9;
    float s = 0.0f;
    for (int c = 0; c < MID; ++c) {
        const float* xc = xp + (size_t)c * NPIX;
        const float* wc = wp + c * 9;
#pragma unroll
        for (int dh = 0; dh < 3; ++dh) {
            const int hh = h + dh - 1;
            if (hh < 0 || hh >= HH) continue;
#pragma unroll
            for (int dw = 0; dw < 3; ++dw) {
                const int ww2 = w_ + dw - 1;
                if (ww2 < 0 || ww2 >= WW) continue;
                s += xc[hh * WW + ww2] * wc[dh * 3 + dw];
            }
        }
    }
    off[idx] = s;
}

// ---------------------------------------------------------------------------
// Kernel 3a: repack dcn_w [O,C,3,3] -> dwt [kpt][O][C] for contiguous A loads
// ---------------------------------------------------------------------------
__global__ __launch_bounds__(256) void cf_repack(
    const float* __restrict__ dw, float* __restrict__ dwt)
{
    const int idx = blockIdx.x * 256 + threadIdx.x;   // KPTS*OUTC*MID
    const int c   = idx % MID;
    const int o   = (idx / MID) % OUTC;
    const int kpt = idx / (MID * OUTC);
    dwt[idx] = dw[(size_t)o * MID * 9 + (size_t)c * 9 + kpt];
}

// ---------------------------------------------------------------------------
// Kernel 3b: deformable conv + BN2.
// Block = 512 thr = 16 waves = all 256 out-ch x 32 pixels. Per kernel-point
// the block materializes sampled[32px][256c] in a double-buffered LDS tile
// (each bilinear gather done once, fill overlaps WMMA of previous tap);
// each wave runs 2 chained-WMMA streams sharing A fragments (128 wmma/tap).
// ---------------------------------------------------------------------------
#define SMPAD (MID + 4)

__global__ __launch_bounds__(512) void cf_dcn(
    const float* __restrict__ xf,  const float* __restrict__ off,
    const float* __restrict__ dwt,
    const float* __restrict__ g2,  const float* __restrict__ b2,
    const float* __restrict__ mu2, const float* __restrict__ var2,
    float* __restrict__ out)
{
    __shared__ float smp[2][32][SMPAD];

    const int tid  = threadIdx.x;
    const int wave = tid >> 5;
    const int lane = tid & 31;
    const int nn   = lane & 15;
    const int hl   = lane >> 4;
    const int nb   = blockIdx.x / (NPIX / 32);
    const int p0   = (blockIdx.x % (NPIX / 32)) * 32;
    const int m0   = wave * 16;

    const float* xp = xf  + (size_t)nb * MID * NPIX;
    const float* op = off + (size_t)nb * 18 * NPIX;

    // fill: thread handles one pixel x 16 channels for kernel-point kpt
    const int fpix = tid & 31;
    const int fcb  = (tid >> 5) * 16;
    const int fp   = p0 + fpix;
    const int fh   = fp / WW, fw = fp % WW;

    auto fill = [&](int kpt, int buf) {
        const int ky = kpt / 3, kx = kpt % 3;
        const float dy = op[(size_t)(2 * kpt)     * NPIX + fp];
        const float dx = op[(size_t)(2 * kpt + 1) * NPIX + fp];
        const float sy = (float)(fh - 1 + ky) + dy;
        const float sx = (float)(fw - 1 + kx) + dx;
        const float fy = floorf(sy), fx = floorf(sx);
        const float wy1 = sy - fy,  wx1 = sx - fx;
        const int y0 = (int)fy, x0 = (int)fx;
        const int y1 = y0 + 1,  x1 = x0 + 1;
        const bool vy0 = (y0 >= 0) && (y0 < HH), vy1 = (y1 >= 0) && (y1 < HH);
        const bool vx0 = (x0 >= 0) && (x0 < WW), vx1 = (x1 >= 0) && (x1 < WW);
        const int yc0 = min(max(y0, 0), HH - 1), yc1 = min(max(y1, 0), HH - 1);
        const int xc0 = min(max(x0, 0), WW - 1), xc1 = min(max(x1, 0), WW - 1);
        const float w00 = (vy0 && vx0) ? (1.0f - wy1) * (1.0f - wx1) : 0.0f;
        const float w01 = (vy0 && vx1) ? (1.0f - wy1) * wx1          : 0.0f;
        const float w10 = (vy1 && vx0) ? wy1 * (1.0f - wx1)          : 0.0f;
        const float w11 = (vy1 && vx1) ? wy1 * wx1                   : 0.0f;
        const int i00 = yc0 * WW + xc0, i01 = yc0 * WW + xc1;
        const int i10 = yc1 * WW + xc0, i11 = yc1 * WW + xc1;

        float* dst = &smp[buf][fpix][fcb];
#pragma unroll
        for (int j = 0; j < 16; ++j) {
            const float* xc = xp + (size_t)(fcb + j) * NPIX;
            dst[j] = w00 * xc[i00] + w01 * xc[i01]
                   + w10 * xc[i10] + w11 * xc[i11];
        }
    };

    fill(0, 0);
    __syncthreads();

    v8f acc0 = {}, acc1 = {};
    for (int kpt = 0; kpt < KPTS; ++kpt) {
        const int buf = kpt & 1;
        if (kpt < KPTS - 1) fill(kpt + 1, buf ^ 1);   // overlap with wmma below

        const float* arow = dwt + (size_t)kpt * OUTC * MID
                                + (size_t)(m0 + nn) * MID;
        const float* b0 = &smp[buf][nn][0];
        const float* b1 = &smp[buf][nn + 16][0];
        for (int cb = 0; cb < MID; cb += 4) {
            const int k = cb + 2 * hl;
            v2f a  = { arow[k], arow[k + 1] };
            v2f bb0 = { b0[k], b0[k + 1] };
            v2f bb1 = { b1[k], b1[k + 1] };
            acc0 = __builtin_amdgcn_wmma_f32_16x16x4_f32(
                false, a, false, bb0, (short)0, acc0, false, false);
            acc1 = __builtin_amdgcn_wmma_f32_16x16x4_f32(
                false, a, false, bb1, (short)0, acc1, false, false);
        }
        __syncthreads();
    }

    // ---- BN2 + store both pixel sets ----
    const int pA = p0 + nn, pB = p0 + nn + 16;
    float* outp = out + (size_t)nb * OUTC * NPIX;
#pragma unroll
    for (int r = 0; r < 8; ++r) {
        const int m  = m0 + r + 8 * hl;
        const float sc = g2[m] * rsqrtf(var2[m] + EPSB);
        const float sh = b2[m] - mu2[m] * sc;
        outp[(size_t)m * NPIX + pA] = acc0[r] * sc + sh;
        outp[(size_t)m * NPIX + pB] = acc1[r] * sc + sh;
    }
}

// ---------------------------------------------------------------------------
extern "C" void kernel_launch(void* const* d_in, const int* in_sizes, int n_in,
                              void* d_out, int out_size, void* d_ws, size_t ws_size,
                              hipStream_t stream) {
    const float* x      = (const float*)d_in[0];
    const float* y      = (const float*)d_in[1];
    const float* conv1w = (const float*)d_in[2];
    const float* bn1g   = (const float*)d_in[3];
    const float* bn1b   = (const float*)d_in[4];
    const float* bn1m   = (const float*)d_in[5];
    const float* bn1v   = (const float*)d_in[6];
    const float* attw   = (const float*)d_in[7];
    const float* attb   = (const float*)d_in[8];
    const float* offw   = (const float*)d_in[9];
    const float* dcnw   = (const float*)d_in[10];
    const float* bn2g   = (const float*)d_in[11];
    const float* bn2b   = (const float*)d_in[12];
    const float* bn2m   = (const float*)d_in[13];
    const float* bn2v   = (const float*)d_in[14];

    float* out = (float*)d_out;                           // output 0: [N,256,56,56]
    float* xf  = out + (size_t)NB * OUTC * NPIX;          // output 1: [N,256,56,56]
    float* off = (float*)d_ws;                            // [N,18,56,56]
    float* dwt = off + (size_t)NB * 18 * NPIX;            // [9,256,256]

    // 1) 1x1 conv + BN1 + upsample + attention fuse -> xf (also output 1)
    cf_fuse1<<<NB * (NPIX / 32), 512, 0, stream>>>(
        x, y, conv1w, bn1g, bn1b, bn1m, bn1v, attw, attb, xf);

    // 2) 3x3 offset conv -> off
    cf_offconv<<<(NB * 18 * NPIX + 255) / 256, 256, 0, stream>>>(xf, offw, off);

    // 3a) repack deformable weights for contiguous A fragments
    cf_repack<<<(KPTS * OUTC * MID) / 256, 256, 0, stream>>>(dcnw, dwt);

    // 3b) deformable conv + BN2 -> out
    cf_dcn<<<NB * (NPIX / 32), 512, 0, stream>>>(
        xf, off, dwt, bn2g, bn2b, bn2m, bn2v, out);
}